// PrototypeContrastiveLoss_64458869178422
// MI455X (gfx1250) — compile-verified
//
#include <hip/hip_runtime.h>
#include <hip/hip_bf16.h>
#include <math.h>

// ---------------- problem constants (fixed by reference setup) ----------------
#define A_DIM   1024
#define C_DIM   6
#define B_DIM   4
#define HW      16384          // 128*128
#define NPIX    65536          // B*H*W
#define PSTRIDE 1028           // padded LDS row stride for Proto (1024 + 4 -> conflict-free)
#define EPSN    1e-12f

typedef __attribute__((ext_vector_type(2))) float v2f;
typedef __attribute__((ext_vector_type(8))) float v8f;

// ---------------------------------------------------------------------------
// Kernel 1: inverse L2 norms of the 6 prototype rows -> stats[12..17]
// ---------------------------------------------------------------------------
__global__ void __launch_bounds__(256)
proto_norm_kernel(const float* __restrict__ Proto, float* __restrict__ stats)
{
    __shared__ float red[256];
    const int c = blockIdx.x;
    const int tid = threadIdx.x;
    float s = 0.f;
    for (int i = tid; i < A_DIM; i += 256) {
        float v = Proto[c * A_DIM + i];
        s += v * v;
    }
    red[tid] = s;
    __syncthreads();
    for (int st = 128; st > 0; st >>= 1) {
        if (tid < st) red[tid] += red[tid + st];
        __syncthreads();
    }
    if (tid == 0) stats[12 + c] = 1.0f / fmaxf(sqrtf(red[0]), EPSN);
}

// ---------------------------------------------------------------------------
// Kernel 2: per-class max and 1/sum(exp) over all N pixels of tarout
//           stats[0..5] = Mc, stats[6..11] = 1/Zc
// ---------------------------------------------------------------------------
__global__ void __launch_bounds__(256)
class_stats_kernel(const float* __restrict__ tarout, float* __restrict__ stats)
{
    __shared__ float red[256];
    const int c = blockIdx.x;
    const int tid = threadIdx.x;

    float m = -__builtin_inff();
    for (int p = tid; p < NPIX; p += 256) {
        const int b = p >> 14, hw = p & (HW - 1);
        m = fmaxf(m, tarout[(size_t)b * C_DIM * HW + (size_t)c * HW + hw]);
    }
    red[tid] = m;
    __syncthreads();
    for (int st = 128; st > 0; st >>= 1) {
        if (tid < st) red[tid] = fmaxf(red[tid], red[tid + st]);
        __syncthreads();
    }
    const float M = red[0];
    __syncthreads();

    float z = 0.f;
    for (int p = tid; p < NPIX; p += 256) {
        const int b = p >> 14, hw = p & (HW - 1);
        z += expf(tarout[(size_t)b * C_DIM * HW + (size_t)c * HW + hw] - M);
    }
    red[tid] = z;
    __syncthreads();
    for (int st = 128; st > 0; st >>= 1) {
        if (tid < st) red[tid] += red[tid + st];
        __syncthreads();
    }
    if (tid == 0) {
        stats[c]     = M;
        stats[6 + c] = 1.0f / red[0];
    }
}

// ---------------------------------------------------------------------------
// Kernel 3: per-class sums + counts of srcfeat (streams 256 MB once).
// Each thread owns 4 columns; 6x4 register accumulators; labels are
// wave-uniform per row so the unrolled compare keeps static indexing.
// ---------------------------------------------------------------------------
__global__ void __launch_bounds__(256)
segmean_kernel(const float* __restrict__ srcfeat, const int* __restrict__ labels,
               float* __restrict__ gsums, float* __restrict__ gcnt)
{
    const int tid = threadIdx.x;
    const int rowBase = blockIdx.x * 128;
    const int col = tid * 4;

    float acc[C_DIM][4];
#pragma unroll
    for (int c = 0; c < C_DIM; ++c)
#pragma unroll
        for (int j = 0; j < 4; ++j) acc[c][j] = 0.f;
    float cnt[C_DIM] = {0.f, 0.f, 0.f, 0.f, 0.f, 0.f};

    for (int r = 0; r < 128; ++r) {
        const int row = rowBase + r;
        const int lab = labels[row];
        if ((unsigned)lab < (unsigned)C_DIM) {   // also masks IGNORE_LABEL=255
            const float4 v = *(const float4*)(srcfeat + (size_t)row * A_DIM + col);
#pragma unroll
            for (int c = 0; c < C_DIM; ++c) {
                if (lab == c) {
                    acc[c][0] += v.x; acc[c][1] += v.y;
                    acc[c][2] += v.z; acc[c][3] += v.w;
                    cnt[c] += 1.f;
                }
            }
        }
    }

#pragma unroll
    for (int c = 0; c < C_DIM; ++c)
#pragma unroll
        for (int j = 0; j < 4; ++j)
            atomicAdd(&gsums[c * A_DIM + col + j], acc[c][j]);

    if (tid == 0) {
#pragma unroll
        for (int c = 0; c < C_DIM; ++c) atomicAdd(&gcnt[c], cnt[c]);
    }
}

// ---------------------------------------------------------------------------
// Kernel 4 (fused, WMMA): cosine contrast + forward & reverse loss partials.
// Per wave32: 16 pixels. A-matrix = padded Proto (16 classes x 4 K) from LDS,
// B-matrix = 16 pixels x 4 K from tarfeat, accumulated over K=1024 with
// v_wmma_f32_16x16x4_f32. D layout: lane n (<16), acc[c] = dot(pixel n, Proto_c).
// ---------------------------------------------------------------------------
__global__ void __launch_bounds__(256)
contrast_loss_kernel(const float* __restrict__ Proto,
                     const float* __restrict__ tarfeat,
                     const float* __restrict__ tarout,
                     const float* __restrict__ stats,   // Mc[6], invZc[6], invPn[6]
                     float* __restrict__ lossAcc)       // [0]=loss1 sum, [1]=loss2 sum
{
    __shared__ float protoLds[16 * PSTRIDE];   // ~64 KB of the 320 KB WGP LDS
    __shared__ float sMc[8], sZc[8], sPn[8];
    __shared__ float blkLoss[2];

    const int tid = threadIdx.x;

    // stage padded prototype matrix (classes 6..15 = 0)
    for (int i = tid; i < 16 * A_DIM; i += 256) {
        const int r = i >> 10, c = i & (A_DIM - 1);
        protoLds[r * PSTRIDE + c] = (r < C_DIM) ? Proto[r * A_DIM + c] : 0.0f;
    }
    if (tid < C_DIM) {
        sMc[tid] = stats[tid];
        sZc[tid] = stats[6 + tid];
        sPn[tid] = stats[12 + tid];
    }
    if (tid == 0) { blkLoss[0] = 0.f; blkLoss[1] = 0.f; }
    __syncthreads();

    const int wave  = tid >> 5;
    const int lane  = tid & 31;
    const int n     = lane & 15;   // pixel within tile / proto row
    const int halfk = lane >> 4;   // K half (0 -> K+0,1 ; 1 -> K+2,3)

    const int gwave   = blockIdx.x * 8 + wave;
    const int pixBase = gwave * 16;                 // 16 consecutive pixels, same batch plane
    const int bIdx    = pixBase >> 14;              // / HW
    const int hwBase  = pixBase & (HW - 1);

    const float* bPtr = tarfeat + (size_t)bIdx * A_DIM * HW + hwBase + n;
    const float* aPtr = protoLds + n * PSTRIDE + 2 * halfk;

    v8f acc = (v8f){0.f, 0.f, 0.f, 0.f, 0.f, 0.f, 0.f, 0.f};
    float sumsq = 0.f;

#pragma unroll 4
    for (int k0 = 0; k0 < A_DIM; k0 += 4) {
        const int k = k0 + 2 * halfk;
        v2f afrag;
        afrag.x = aPtr[k0];
        afrag.y = aPtr[k0 + 1];
        const float b0 = bPtr[(size_t)k * HW];
        const float b1 = bPtr[(size_t)(k + 1) * HW];
        v2f bfrag;
        bfrag.x = b0;
        bfrag.y = b1;
        sumsq += b0 * b0 + b1 * b1;
        acc = __builtin_amdgcn_wmma_f32_16x16x4_f32(
            /*neg_a=*/false, afrag, /*neg_b=*/false, bfrag,
            /*c_mod=*/(short)0, acc, /*reuse_a=*/false, /*reuse_b=*/false);
    }

    // complete per-pixel ||x||^2 : lanes L and L^16 each hold half the K range
    sumsq += __shfl_xor(sumsq, 16, 32);

    float l1 = 0.f, l2 = 0.f;
    if (lane < 16) {
        const float invXn = 1.0f / fmaxf(sqrtf(sumsq), EPSN);
        const int p  = pixBase + lane;
        const int b2 = p >> 14;
        const int hw = p & (HW - 1);
        const float* tptr = tarout + (size_t)b2 * C_DIM * HW + hw;

        float t[C_DIM], bb[C_DIM];
        float m = -__builtin_inff();
#pragma unroll
        for (int c = 0; c < C_DIM; ++c) {
            t[c] = tptr[(size_t)c * HW];
            m = fmaxf(m, t[c]);
            const float contrast = acc[c] * sPn[c] * invXn;
            bb[c] = 1.0f - contrast;
        }
        float Z = 0.f, s1 = 0.f;
#pragma unroll
        for (int c = 0; c < C_DIM; ++c) {
            const float e = expf(t[c] - m);
            Z  += e;
            s1 += e * bb[c];
            l2 += expf(t[c] - sMc[c]) * sZc[c] * bb[c];   // sZc = 1/Z_c
        }
        l1 = s1 / Z;
    }

    atomicAdd(&blkLoss[0], l1);
    atomicAdd(&blkLoss[1], l2);
    __syncthreads();
    if (tid == 0) {
        atomicAdd(&lossAcc[0], blkLoss[0]);
        atomicAdd(&lossAcc[1], blkLoss[1]);
    }
}

// ---------------------------------------------------------------------------
// Kernel 5: finalize -> d_out[0] = loss, d_out[1..6144] = mean (C,A)
// ---------------------------------------------------------------------------
__global__ void __launch_bounds__(256)
finalize_kernel(const float* __restrict__ gsums, const float* __restrict__ gcnt,
                const float* __restrict__ lossAcc, float* __restrict__ out)
{
    const int i = blockIdx.x * blockDim.x + threadIdx.x;
    if (i < C_DIM * A_DIM) {
        const int c = i >> 10;
        out[1 + i] = gsums[i] / fmaxf(gcnt[c], 1.0f);
    }
    if (i == 0)
        out[0] = lossAcc[0] / (float)NPIX + lossAcc[1] / (float)C_DIM;
}

// ---------------------------------------------------------------------------
extern "C" void kernel_launch(void* const* d_in, const int* in_sizes, int n_in,
                              void* d_out, int out_size, void* d_ws, size_t ws_size,
                              hipStream_t stream)
{
    const float* Proto   = (const float*)d_in[0];   // (6, 1024)
    const float* srcfeat = (const float*)d_in[1];   // (65536, 1024)
    const float* tarfeat = (const float*)d_in[2];   // (4, 1024, 128, 128)
    const float* tarout  = (const float*)d_in[3];   // (4, 6, 128, 128)
    const int*   labels  = (const int*)d_in[4];     // (65536,)
    float* out = (float*)d_out;                     // [loss, mean(6x1024)] = 6145 floats

    float* ws      = (float*)d_ws;
    float* gsums   = ws;            // 6144 : per-class feature sums
    float* gcnt    = ws + 6144;     // 6    : per-class counts
    float* stats   = ws + 6150;     // 18   : Mc[6], invZc[6], invPnorm[6]
    float* lossAcc = ws + 6168;     // 2    : loss1 sum, loss2 sum

    hipMemsetAsync(d_ws, 0, 6170 * sizeof(float), stream);

    proto_norm_kernel  <<<C_DIM, 256, 0, stream>>>(Proto, stats);
    class_stats_kernel <<<C_DIM, 256, 0, stream>>>(tarout, stats);
    segmean_kernel     <<<NPIX / 128, 256, 0, stream>>>(srcfeat, labels, gsums, gcnt);
    contrast_loss_kernel<<<NPIX / 128, 256, 0, stream>>>(Proto, tarfeat, tarout, stats, lossAcc);
    finalize_kernel    <<<(C_DIM * A_DIM + 255) / 256, 256, 0, stream>>>(gsums, gcnt, lossAcc, out);
}